// GTN_84550726189101
// MI455X (gfx1250) — compile-verified
//
#include <hip/hip_runtime.h>
#include <hip/hip_bf16.h>

// ---------------------------------------------------------------------------
// GTN forward for MI455X (gfx1250).
//  - Heavy GEMMs: V_WMMA_F32_16X16X4_F32 (CDNA5 fp32 matrix op).
//  - Tile staging: GLOBAL_LOAD_ASYNC_TO_LDS_B128 (contiguous A tiles) and
//    GLOBAL_LOAD_ASYNC_TO_LDS_B32 (transposing B / trans-A tiles) -- ASYNCcnt
//    DMA, no VGPR staging, no scratch spills; double-buffered LDS.
//  - All fragments are aligned float2 LDS loads -> ds_load_2addr_b64 straight
//    into WMMA operand pairs (no v_mov shuffles).
// ---------------------------------------------------------------------------

typedef __attribute__((ext_vector_type(2))) float v2f;
typedef __attribute__((ext_vector_type(8))) float v8f;

#define GTN_E 5
#define GTN_C 2
#define GTN_N 2048
#define GTN_WIN 256
#define GTN_WOUT 64
#define GTN_NCLS 8

#define BK 16
#define PAD 20   // padded LDS row (floats): 80B rows, 16B-aligned quads, 8B frag loads

__device__ __forceinline__ void async_to_lds_b128(const float* g, unsigned lds_off) {
  asm volatile("global_load_async_to_lds_b128 %0, %1, off" :: "v"(lds_off), "v"(g) : "memory");
}
__device__ __forceinline__ void async_to_lds_b32(const float* g, unsigned lds_off) {
  asm volatile("global_load_async_to_lds_b32 %0, %1, off" :: "v"(lds_off), "v"(g) : "memory");
}
__device__ __forceinline__ void wait_async0() {
#if __has_builtin(__builtin_amdgcn_s_wait_asynccnt)
  __builtin_amdgcn_s_wait_asynccnt(0);
#else
  asm volatile("s_wait_asynccnt 0x0" ::: "memory");
#endif
}

// -------------------------- small helper kernels ---------------------------

__global__ void k_softmax(const float* __restrict__ w1a, const float* __restrict__ w1b,
                          const float* __restrict__ w2, float* __restrict__ f) {
  int t = threadIdx.x;
  if (t >= 3 * GTN_C) return;
  const float* src = (t < GTN_C) ? (w1a + t * GTN_E)
                   : (t < 2 * GTN_C) ? (w1b + (t - GTN_C) * GTN_E)
                   : (w2 + (t - 2 * GTN_C) * GTN_E);
  float m = -3.0e38f;
  for (int e = 0; e < GTN_E; ++e) m = fmaxf(m, src[e]);
  float tmp[GTN_E]; float s = 0.f;
  for (int e = 0; e < GTN_E; ++e) { tmp[e] = __expf(src[e] - m); s += tmp[e]; }
  float inv = 1.f / s;
  for (int e = 0; e < GTN_E; ++e) f[t * GTN_E + e] = tmp[e] * inv;
}

__global__ void k_qmix(const float* __restrict__ A, const float* __restrict__ fa,
                       const float* __restrict__ fb, float* __restrict__ Qa,
                       float* __restrict__ Qb, long nn) {
  long p = ((long)blockIdx.x * blockDim.x + threadIdx.x) * 4;
  if (p >= nn) return;
  float4 a[GTN_E];
#pragma unroll
  for (int e = 0; e < GTN_E; ++e) a[e] = *(const float4*)(A + (long)e * nn + p);
#pragma unroll
  for (int c = 0; c < GTN_C; ++c) {
    float4 qa = make_float4(0.f, 0.f, 0.f, 0.f);
#pragma unroll
    for (int e = 0; e < GTN_E; ++e) {
      float w = fa[c * GTN_E + e];
      qa.x += w * a[e].x; qa.y += w * a[e].y; qa.z += w * a[e].z; qa.w += w * a[e].w;
    }
    *(float4*)(Qa + (long)c * nn + p) = qa;
    if (Qb) {
      float4 qb = make_float4(0.f, 0.f, 0.f, 0.f);
#pragma unroll
      for (int e = 0; e < GTN_E; ++e) {
        float w = fb[c * GTN_E + e];
        qb.x += w * a[e].x; qb.y += w * a[e].y; qb.z += w * a[e].z; qb.w += w * a[e].w;
      }
      *(float4*)(Qb + (long)c * nn + p) = qb;
    }
  }
}

// ------------------------- WMMA fp32 GEMM kernel ---------------------------
// D[M,N] = opA(A)[M,K] @ B[K,N], row-major; TRANSA => A stored [K,M].
// Compile-time epilogue: D = relu( D*rowScale[m] + bias[n] ).
// Batched over blockIdx.z with element strides.
// LDS: As[m][kk], Bs[n][kk], rows padded to PAD=20 floats; async DMA fills
// them (b128 for contiguous A rows, transposing b32 for B / trans-A).
template <int BMT, int BNT, int WAVES_M, int WAVES_N, int WMT, int WNT,
          bool TRANSA, bool HAS_RS, bool HAS_BIAS, bool RELU>
__global__ __launch_bounds__(WAVES_M * WAVES_N * 32, 1)
void k_gemm_wmma(const float* __restrict__ A, const float* __restrict__ B,
                 float* __restrict__ D, int M, int N, int K, int lda, int ldb, int ldd,
                 long strideA, long strideB, long strideD,
                 const float* __restrict__ rowScale, long strideRS,
                 const float* __restrict__ bias) {
  constexpr int THREADS = WAVES_M * WAVES_N * 32;
  constexpr int A128 = (BMT * BK / 4) / THREADS;   // b128 copies per thread (no-trans A)
  constexpr int A32  = (BMT * BK) / THREADS;       // b32 copies per thread (trans A)
  constexpr int B32  = (BNT * BK) / THREADS;       // b32 copies per thread (B)
  static_assert(A128 * THREADS == BMT * BK / 4, "A tile mismatch");
  static_assert(B32 * THREADS == BNT * BK, "B tile mismatch");

  __shared__ __align__(16) float As[2][BMT * PAD];
  __shared__ __align__(16) float Bs[2][BNT * PAD];

  const int tid  = threadIdx.x;
  const int lane = tid & 31;
  const int w    = tid >> 5;
  const int wmo  = (w / WAVES_N) * (WMT * 16);
  const int wno  = (w % WAVES_N) * (WNT * 16);
  const int ln   = lane & 15;
  const int hi   = lane >> 4;           // lane-hi selects K+2 pair (ISA fp32 16x4 layout)

  const int row0 = blockIdx.y * BMT;
  const int col0 = blockIdx.x * BNT;
  const int z    = blockIdx.z;
  A += (long)z * strideA;
  B += (long)z * strideB;
  D += (long)z * strideD;

  v8f acc[WMT][WNT] = {};

  auto asyncA = [&](int k0, int nb) {
    if constexpr (!TRANSA) {
#pragma unroll
      for (int i = 0; i < A128; ++i) {
        int s = tid + i * THREADS;
        int m = s >> 2, kq = (s & 3) * 4;
        async_to_lds_b128(A + (long)(row0 + m) * lda + (k0 + kq),
                          (unsigned)(unsigned long)&As[nb][m * PAD + kq]);
      }
    } else {
#pragma unroll
      for (int i = 0; i < A32; ++i) {
        int s = tid + i * THREADS;
        int m = s % BMT, kk = s / BMT;   // lanes -> consecutive m: coalesced global reads
        async_to_lds_b32(A + (long)(k0 + kk) * lda + (row0 + m),
                         (unsigned)(unsigned long)&As[nb][m * PAD + kk]);
      }
    }
  };
  auto asyncB = [&](int k0, int nb) {
#pragma unroll
    for (int i = 0; i < B32; ++i) {
      int s = tid + i * THREADS;
      int n = s % BNT, kk = s / BNT;     // lanes -> consecutive n: coalesced global reads
      async_to_lds_b32(B + (long)(k0 + kk) * ldb + (col0 + n),
                       (unsigned)(unsigned long)&Bs[nb][n * PAD + kk]);
    }
  };
  auto compute = [&](int buf) {
#pragma unroll
    for (int ks = 0; ks < BK; ks += 4) {
      const int kr = ks + 2 * hi;
      v2f af[WMT], bf[WNT];
#pragma unroll
      for (int mi = 0; mi < WMT; ++mi)
        af[mi] = *(const v2f*)(&As[buf][(wmo + mi * 16 + ln) * PAD + kr]);
#pragma unroll
      for (int ni = 0; ni < WNT; ++ni)
        bf[ni] = *(const v2f*)(&Bs[buf][(wno + ni * 16 + ln) * PAD + kr]);
#pragma unroll
      for (int mi = 0; mi < WMT; ++mi)
#pragma unroll
        for (int ni = 0; ni < WNT; ++ni)
          acc[mi][ni] = __builtin_amdgcn_wmma_f32_16x16x4_f32(
              false, af[mi], false, bf[ni], (short)0, acc[mi][ni], false, false);
    }
  };

  // prologue: tile 0 -> buffer 0 via async DMA
  asyncA(0, 0); asyncB(0, 0);
  wait_async0();
  __syncthreads();

  const int nIter = K / BK;
  for (int it = 0; it < nIter; ++it) {
    const int cur = it & 1;
    const bool more = (it + 1) < nIter;
    if (more) { asyncA((it + 1) * BK, cur ^ 1); asyncB((it + 1) * BK, cur ^ 1); }
    compute(cur);                         // 32 WMMAs hide the async DMA
    if (more) wait_async0();              // own async copies done
    __syncthreads();                      // everyone's copies visible
  }

  const float* rs = HAS_RS ? (rowScale + (long)z * strideRS) : nullptr;
#pragma unroll
  for (int mi = 0; mi < WMT; ++mi) {
#pragma unroll
    for (int r = 0; r < 8; ++r) {        // C/D layout: row = r + 8*hi, col = lane&15
      int rowg = row0 + wmo + mi * 16 + r + 8 * hi;
      float rsv = 1.f;
      if constexpr (HAS_RS) rsv = rs[rowg];
#pragma unroll
      for (int ni = 0; ni < WNT; ++ni) {
        int colg = col0 + wno + ni * 16 + ln;
        float v = acc[mi][ni][r];
        if constexpr (HAS_RS)   v *= rsv;
        if constexpr (HAS_BIAS) v += bias[colg];
        if constexpr (RELU)     v = fmaxf(v, 0.f);
        D[(long)rowg * ldd + colg] = v;
      }
    }
  }
}

// -------------------- deterministic reductions / scales --------------------

template <int MODE>   // 0: sum values, 1: count nonzeros
__global__ void k_colreduce(const float* __restrict__ H, float* __restrict__ partial,
                            int nrows, int ncols, long batchStride, int chunks) {
  int c = blockIdx.z, chunk = blockIdx.y;
  int col = blockIdx.x * blockDim.x + threadIdx.x;
  int rpc = nrows / chunks;
  int r0 = chunk * rpc;
  const float* Hc = H + (long)c * batchStride;
  float s = 0.f;
  for (int i = r0; i < r0 + rpc; ++i) {
    float v = Hc[(long)i * ncols + col];
    s += (MODE == 0) ? v : (v != 0.f ? 1.f : 0.f);
  }
  partial[((long)chunk * gridDim.z + c) * ncols + col] = s;
}

__global__ void k_colcombine(const float* __restrict__ partial, float* __restrict__ out,
                             int ncols, int Cb, int chunks) {
  int idx = blockIdx.x * blockDim.x + threadIdx.x;
  if (idx >= Cb * ncols) return;
  int c = idx / ncols, col = idx % ncols;
  float s = 0.f;
  for (int ch = 0; ch < chunks; ++ch) s += partial[((long)ch * Cb + c) * ncols + col];
  out[idx] = s;
}

__global__ void k_rowcount(const float* __restrict__ H, float* __restrict__ out,
                           int n, long batchStride) {
  __shared__ float red[256];
  int c = blockIdx.y, row = blockIdx.x;
  const float* r = H + (long)c * batchStride + (long)row * n;
  float s = 0.f;
  for (int j = threadIdx.x; j < n; j += 256) s += (r[j] != 0.f) ? 1.f : 0.f;
  red[threadIdx.x] = s;
  __syncthreads();
  for (int o = 128; o > 0; o >>= 1) {
    if (threadIdx.x < o) red[threadIdx.x] += red[threadIdx.x + o];
    __syncthreads();
  }
  if (threadIdx.x == 0) out[(long)c * n + row] = red[0];
}

// mode 0: x = (x!=0) ? 1/x : 0 ; mode 1: x = rsqrt(max(x,1))
__global__ void k_transform(float* __restrict__ x, long n, int mode) {
  long i = (long)blockIdx.x * blockDim.x + threadIdx.x;
  if (i >= n) return;
  float v = x[i];
  x[i] = mode ? rsqrtf(fmaxf(v, 1.f)) : (v != 0.f ? 1.f / v : 0.f);
}

__global__ void k_colscale(float* __restrict__ H, const float* __restrict__ dinv,
                           int ncols, long batchStride, long nn) {
  int c = blockIdx.y;
  long p = ((long)blockIdx.x * blockDim.x + threadIdx.x) * 4;
  if (p >= nn) return;
  int col = (int)(p % (long)ncols);
  const float* di = dinv + (long)c * ncols;
  float4* Hp = (float4*)(H + (long)c * batchStride + p);
  float4 v = *Hp;
  v.x *= di[col]; v.y *= di[col + 1]; v.z *= di[col + 2]; v.w *= di[col + 3];
  *Hp = v;
}

__global__ void k_feat(const float* __restrict__ h, const float* __restrict__ srcn,
                       float* __restrict__ feat, int n, int wdim) {
  long p = ((long)blockIdx.x * blockDim.x + threadIdx.x) * 4;
  long per = (long)n * wdim;
  if (p >= (long)GTN_C * per) return;
  int c = (int)(p / per);
  long rem = p - (long)c * per;
  int i = (int)(rem / wdim);
  float s = srcn[(long)c * n + i];
  float4 v = *(const float4*)(h + rem);
  v.x *= s; v.y *= s; v.z *= s; v.w *= s;
  *(float4*)(feat + p) = v;
}

__global__ void k_head(const float* __restrict__ X1, const float* __restrict__ w,
                       const float* __restrict__ b, float* __restrict__ y,
                       int n, int k, int ncls) {
  int idx = blockIdx.x * blockDim.x + threadIdx.x;
  if (idx >= n * ncls) return;
  int node = idx / ncls, cl = idx % ncls;
  float s = b[cl];
  for (int j = 0; j < k; ++j) s += X1[(long)node * k + j] * w[j * ncls + cl];
  y[idx] = s;
}

// ------------------------------- launcher ----------------------------------

extern "C" void kernel_launch(void* const* d_in, const int* in_sizes, int n_in,
                              void* d_out, int out_size, void* d_ws, size_t ws_size,
                              hipStream_t stream) {
  (void)in_sizes; (void)n_in; (void)out_size; (void)ws_size;
  constexpr int  E = GTN_E, C = GTN_C, N = GTN_N, WIN = GTN_WIN, WOUT = GTN_WOUT, NCLS = GTN_NCLS;
  constexpr long NNl = (long)N * N;
  constexpr long CNN = (long)C * NNl;
  constexpr int  CHUNKS = 16;

  const float* A      = (const float*)d_in[0];
  const float* h      = (const float*)d_in[1];
  const float* gt_w1a = (const float*)d_in[2];
  const float* gt_w1b = (const float*)d_in[3];
  const float* gt_w2  = (const float*)d_in[4];
  const float* gcn_w  = (const float*)d_in[5];
  const float* gcn_b  = (const float*)d_in[6];
  const float* lin1_w = (const float*)d_in[7];
  const float* lin1_b = (const float*)d_in[8];
  const float* lin2_w = (const float*)d_in[9];
  const float* lin2_b = (const float*)d_in[10];
  float* y = (float*)d_out;

  float* W = (float*)d_ws;
  float* buf0 = W;                           // C*N*N
  float* buf1 = buf0 + CNN;                  // C*N*N
  float* buf2 = buf1 + CNN;                  // C*N*N
  float* deg  = buf2 + CNN;                  // C*N
  float* srcn = deg  + (long)C * N;          // C*N
  float* dstn = srcn + (long)C * N;          // C*N
  float* part = dstn + (long)C * N;          // CHUNKS*C*N
  float* fsm  = part + (long)CHUNKS * C * N; // 3*C*E (<=32)
  float* feat = fsm + 32;                    // C*N*WIN
  float* rst  = feat + (long)C * N * WIN;    // C*N*WIN
  float* Xc   = rst  + (long)C * N * WIN;    // N*(C*WOUT)
  float* X1   = Xc   + (long)N * C * WOUT;   // N*WOUT

  const dim3 blk256(256);
  const dim3 gq((unsigned)(NNl / 4 / 256));

  // softmax of edge-type mixing weights
  k_softmax<<<1, 32, 0, stream>>>(gt_w1a, gt_w1b, gt_w2, fsm);

  // Q1 (buf0), Q2 (buf1)
  k_qmix<<<gq, blk256, 0, stream>>>(A, fsm, fsm + C * E, buf0, buf1, NNl);

  // H1 (buf2) = Q1 @ Q2, per channel  -- 128x128 tiles, 8 waves
  k_gemm_wmma<128, 128, 4, 2, 2, 4, false, false, false, false>
      <<<dim3(N / 128, N / 128, C), 256, 0, stream>>>(
      buf0, buf1, buf2, N, N, N, N, N, N, NNl, NNl, NNl, nullptr, 0, nullptr);

  // normalize H1 by column sums
  k_colreduce<0><<<dim3(N / 256, CHUNKS, C), blk256, 0, stream>>>(buf2, part, N, N, NNl, CHUNKS);
  k_colcombine<<<dim3((C * N + 255) / 256), blk256, 0, stream>>>(part, deg, N, C, CHUNKS);
  k_transform<<<dim3((C * N + 255) / 256), blk256, 0, stream>>>(deg, (long)C * N, 0);
  k_colscale<<<dim3((unsigned)(NNl / 4 / 256), C), blk256, 0, stream>>>(buf2, deg, N, NNl, NNl);

  // Q (buf0, overwrite Q1)
  k_qmix<<<gq, blk256, 0, stream>>>(A, fsm + 2 * C * E, nullptr, buf0, nullptr, NNl);

  // H2 (buf1, overwrite Q2) = H1n @ Q
  k_gemm_wmma<128, 128, 4, 2, 2, 4, false, false, false, false>
      <<<dim3(N / 128, N / 128, C), 256, 0, stream>>>(
      buf2, buf0, buf1, N, N, N, N, N, N, NNl, NNl, NNl, nullptr, 0, nullptr);

  // normalize H2 by column sums
  k_colreduce<0><<<dim3(N / 256, CHUNKS, C), blk256, 0, stream>>>(buf1, part, N, N, NNl, CHUNKS);
  k_colcombine<<<dim3((C * N + 255) / 256), blk256, 0, stream>>>(part, deg, N, C, CHUNKS);
  k_transform<<<dim3((C * N + 255) / 256), blk256, 0, stream>>>(deg, (long)C * N, 0);
  k_colscale<<<dim3((unsigned)(NNl / 4 / 256), C), blk256, 0, stream>>>(buf1, deg, N, NNl, NNl);

  // GCN degree norms (unweighted nnz counts on H2n)
  k_rowcount<<<dim3(N, C), blk256, 0, stream>>>(buf1, srcn, N, NNl);
  k_transform<<<dim3((C * N + 255) / 256), blk256, 0, stream>>>(srcn, (long)C * N, 1);
  k_colreduce<1><<<dim3(N / 256, CHUNKS, C), blk256, 0, stream>>>(buf1, part, N, N, NNl, CHUNKS);
  k_colcombine<<<dim3((C * N + 255) / 256), blk256, 0, stream>>>(part, dstn, N, C, CHUNKS);
  k_transform<<<dim3((C * N + 255) / 256), blk256, 0, stream>>>(dstn, (long)C * N, 1);

  // feat[c] = h * src_norm[c]
  k_feat<<<dim3((unsigned)((long)C * N * WIN / 4 / 256)), blk256, 0, stream>>>(h, srcn, feat, N, WIN);

  // rst[c] = H2n[c]^T @ feat[c], fused dst_norm row scale (M=N, cols=WIN, K=N)
  k_gemm_wmma<128, 128, 4, 2, 2, 4, true, true, false, false>
      <<<dim3(WIN / 128, N / 128, C), 256, 0, stream>>>(
      buf1, feat, rst, N, WIN, N, N, WIN, WIN,
      NNl, (long)N * WIN, (long)N * WIN, dstn, N, nullptr);

  // Xc[:, c*64:(c+1)*64] = relu(rst[c] @ gcn_w + gcn_b)  (strideD=64 writes concat)
  k_gemm_wmma<64, 64, 2, 2, 2, 2, false, false, true, true>
      <<<dim3(WOUT / 64, N / 64, C), 128, 0, stream>>>(
      rst, gcn_w, Xc, N, WOUT, WIN, WIN, WOUT, C * WOUT,
      (long)N * WIN, 0, (long)WOUT, nullptr, 0, gcn_b);

  // X1 = relu(Xc @ lin1_w + lin1_b)
  k_gemm_wmma<64, 64, 2, 2, 2, 2, false, false, true, true>
      <<<dim3(WOUT / 64, N / 64, 1), 128, 0, stream>>>(
      Xc, lin1_w, X1, N, WOUT, C * WOUT, C * WOUT, WOUT, WOUT,
      0, 0, 0, nullptr, 0, lin1_b);

  // y = X1 @ lin2_w + lin2_b
  k_head<<<dim3((N * NCLS + 255) / 256), blk256, 0, stream>>>(X1, lin2_w, lin2_b, y, N, WOUT, NCLS);
}